// HybridTransformer_65481071410033
// MI455X (gfx1250) — compile-verified
//
#include <hip/hip_runtime.h>

// ---- problem constants (match reference) ----
#define Bc   4
#define Tc   2048
#define Ec   16
#define Hc   4
#define Dc   4
#define Lc   2
#define FFc  64
#define Cc   16
#define Nc   (Bc * Tc)      // 8192 token rows
#define NEc  (Nc * Ec)      // 131072 floats per activation buffer
#define KT   (Tc / 16)      // 128 key tiles per (b,h)

typedef __attribute__((ext_vector_type(2))) float v2f;
typedef __attribute__((ext_vector_type(8))) float v8f;
typedef __attribute__((ext_vector_type(4))) unsigned vu4;
typedef __attribute__((ext_vector_type(8))) int vi8;
typedef __attribute__((ext_vector_type(4))) int vi4;

// D = A(16x4) * B(4x16) + C, fp32, one CDNA5 WMMA.
// A/B frag: lane (half = lane>>4, n = lane&15), element v -> K = v + 2*half.
// C/D frag: element r -> row = r + 8*half, col = n.
__device__ __forceinline__ v8f wmma4(v2f a, v2f b, v8f c) {
  return __builtin_amdgcn_wmma_f32_16x16x4_f32(false, a, false, b, (short)0, c,
                                               false, false);
}

// ---------------- kernel 1: embedding + sinusoidal PE ----------------
__global__ void k_embed(const int* __restrict__ tokens,
                        const float* __restrict__ emb, float* __restrict__ x) {
  int idx = blockIdx.x * blockDim.x + threadIdx.x;
  if (idx >= NEc) return;
  int row = idx >> 4;          // 0..N-1
  int e   = idx & 15;
  int t   = row & (Tc - 1);    // position within sequence
  int tok = tokens[row];
  float i2  = (float)(e & ~1);
  float ang = (float)t * __expf(-logf(10000.f) * i2 / (float)Ec);
  float pe  = (e & 1) ? __cosf(ang) : __sinf(ang);
  x[idx] = emb[tok * Ec + e] + pe;
}

// ---------------- kernel 2: fused Q/K/V projection (x @ W, K=16) ----------------
__global__ void k_qkv(const float* __restrict__ x, const float* __restrict__ Wq,
                      const float* __restrict__ Wk, const float* __restrict__ Wv,
                      float* __restrict__ q, float* __restrict__ k,
                      float* __restrict__ v) {
  const int wave = (blockIdx.x * blockDim.x + threadIdx.x) >> 5;  // 0..511
  const int lane = threadIdx.x & 31;
  const int half = lane >> 4, n = lane & 15;
  const int row0 = wave << 4;

  v2f a[4];  // x tile, A layout, K-steps of 4
#pragma unroll
  for (int kk = 0; kk < 4; ++kk) {
    const float* p = x + (row0 + n) * Ec + kk * 4 + 2 * half;
    a[kk][0] = p[0];
    a[kk][1] = p[1];
  }
  const float* Ws[3] = {Wq, Wk, Wv};
  float* outs[3] = {q, k, v};
#pragma unroll
  for (int w = 0; w < 3; ++w) {
    const float* __restrict__ W = Ws[w];
    v8f c = {};
#pragma unroll
    for (int kk = 0; kk < 4; ++kk) {
      v2f bf;
      bf[0] = W[(kk * 4 + 2 * half + 0) * Ec + n];
      bf[1] = W[(kk * 4 + 2 * half + 1) * Ec + n];
      c = wmma4(a[kk], bf, c);
    }
    float* __restrict__ o = outs[w];
#pragma unroll
    for (int r = 0; r < 8; ++r) o[(row0 + r + 8 * half) * Ec + n] = c[r];
  }
}

// ---------------- kernel 3: attention ----------------
// One block per (b,h, group-of-32 qtiles): 32 waves, TDM stages the K and V
// head-slice panels (2048x4 f32, row stride 16 f32) into LDS, then each wave
// runs a two-pass (exact-max) softmax over its 16 query rows with one
// v_wmma_f32_16x16x4_f32 per 16-key tile per pass.
__global__ void __launch_bounds__(1024) k_attn(const float* __restrict__ q,
                                               const float* __restrict__ k,
                                               const float* __restrict__ v,
                                               float* __restrict__ ao) {
  __shared__ float sh[16384];  // [0,8192): K panel [2048][4]; [8192,16384): V panel
  const int tid  = threadIdx.x;
  const int lane = tid & 31;
  const int half = lane >> 4, n = lane & 15;
  const int wv   = tid >> 5;        // 0..31
  const int blk  = blockIdx.x;      // 0..63
  const int bh   = blk >> 2;        // 0..15
  const int h    = bh & (Hc - 1);
  const int b    = bh >> 2;
  const int qt   = (blk & 3) * 32 + wv;   // 0..127
  const int qrow0 = b * Tc + qt * 16;
  const int krow0 = b * Tc;

  // ---- TDM: DMA K/V panels (strided 2D tiles) into LDS; wave 0 issues ----
  if (wv == 0) {
    unsigned long long gk =
        (unsigned long long)(size_t)(const void*)(k + (size_t)krow0 * Ec + h * Dc);
    unsigned long long gv =
        (unsigned long long)(size_t)(const void*)(v + (size_t)krow0 * Ec + h * Dc);
    // D# group0: count=1 | lds_addr | global_addr[56:0] | type=2
    vu4 g0k, g0v;
    g0k[0] = 1u; g0k[1] = 0u;
    g0k[2] = (unsigned)gk;
    g0k[3] = (unsigned)((gk >> 32) & 0x01FFFFFFu) | 0x80000000u;
    g0v[0] = 1u; g0v[1] = 32768u;   // V panel at LDS byte offset 32KB
    g0v[2] = (unsigned)gv;
    g0v[3] = (unsigned)((gv >> 32) & 0x01FFFFFFu) | 0x80000000u;
    // D# group1: wg_mask=0, data_size=4B, tensor_dim=[4,2048],
    //            tile_dim=[4,2048], tensor_dim0_stride=16
    vi8 g1;
    g1[0] = 0x20000;                 // data_size = 2 (4 bytes)
    g1[1] = (int)(4u << 16);         // tensor_dim0 = 4
    g1[2] = (int)(2048u << 16);      // tensor_dim1 = 2048
    g1[3] = (int)(4u << 16);         // tile_dim0 = 4
    g1[4] = 2048;                    // tile_dim1 = 2048
    g1[5] = 16;                      // tensor_dim0_stride = 16 elements
    g1[6] = 0;
    g1[7] = 0;
    vi4 z4 = {0, 0, 0, 0};
    vi8 z8 = {0, 0, 0, 0, 0, 0, 0, 0};
    __builtin_amdgcn_tensor_load_to_lds(g0k, g1, z4, z4, z8, 0);
    __builtin_amdgcn_tensor_load_to_lds(g0v, g1, z4, z4, z8, 0);
    __builtin_amdgcn_s_wait_tensorcnt(0);
  }
  __syncthreads();
  const float* kl = sh;          // [2048][4]
  const float* vl = sh + 8192;   // [2048][4]

  // Q tile as A frag, pre-scaled by 1/sqrt(D) = 0.5
  v2f a;
  {
    const float* p = q + (qrow0 + n) * Ec + h * Dc + 2 * half;
    a[0] = p[0] * 0.5f;
    a[1] = p[1] * 0.5f;
  }

  // ---- pass 1: exact row max (per-lane max in loop; one reduction after) ----
  float mx[8];
#pragma unroll
  for (int r = 0; r < 8; ++r) mx[r] = -3.0e38f;
  for (int kt = 0; kt < KT; ++kt) {
    const float* kp = kl + (kt * 16 + n) * 4 + 2 * half;
    v2f bf;
    bf[0] = kp[0];
    bf[1] = kp[1];
    v8f s = {};
    s = wmma4(a, bf, s);
#pragma unroll
    for (int r = 0; r < 8; ++r) mx[r] = fmaxf(mx[r], s[r]);
  }
#pragma unroll
  for (int r = 0; r < 8; ++r) {
#pragma unroll
    for (int m = 1; m <= 8; m <<= 1) mx[r] = fmaxf(mx[r], __shfl_xor(mx[r], m));
  }

  // ---- pass 2: exp/accumulate with deferred cross-lane sums ----
  float lsum[8], oacc[8][4];
#pragma unroll
  for (int r = 0; r < 8; ++r) {
    lsum[r] = 0.f;
    oacc[r][0] = oacc[r][1] = oacc[r][2] = oacc[r][3] = 0.f;
  }
  for (int kt = 0; kt < KT; ++kt) {
    const float* kp = kl + (kt * 16 + n) * 4 + 2 * half;
    v2f bf;
    bf[0] = kp[0];
    bf[1] = kp[1];
    float4 vv = *(const float4*)(vl + (kt * 16 + n) * 4);
    v8f s = {};
    s = wmma4(a, bf, s);
#pragma unroll
    for (int r = 0; r < 8; ++r) {
      float p = __expf(s[r] - mx[r]);
      lsum[r] += p;
      oacc[r][0] += p * vv.x;
      oacc[r][1] += p * vv.y;
      oacc[r][2] += p * vv.z;
      oacc[r][3] += p * vv.w;
    }
  }

  // one-time cross-lane reduction, normalize, write O
#pragma unroll
  for (int r = 0; r < 8; ++r) {
    float ls = lsum[r];
    float o0 = oacc[r][0], o1 = oacc[r][1], o2 = oacc[r][2], o3 = oacc[r][3];
#pragma unroll
    for (int m = 1; m <= 8; m <<= 1) {
      ls += __shfl_xor(ls, m);
      o0 += __shfl_xor(o0, m);
      o1 += __shfl_xor(o1, m);
      o2 += __shfl_xor(o2, m);
      o3 += __shfl_xor(o3, m);
    }
    if (n == 0) {
      float inv = 1.f / ls;
      float4 o4 = make_float4(o0 * inv, o1 * inv, o2 * inv, o3 * inv);
      *(float4*)(ao + (qrow0 + r + 8 * half) * Ec + h * Dc) = o4;
    }
  }
}

// ---------------- kernel 4: output projection + residual ----------------
__global__ void k_oproj(const float* __restrict__ ao,
                        const float* __restrict__ Wo, float* __restrict__ x) {
  const int wave = (blockIdx.x * blockDim.x + threadIdx.x) >> 5;
  const int lane = threadIdx.x & 31;
  const int half = lane >> 4, n = lane & 15;
  const int row0 = wave << 4;

  v8f c = {};
#pragma unroll
  for (int kk = 0; kk < 4; ++kk) {
    v2f af;
    const float* p = ao + (row0 + n) * Ec + kk * 4 + 2 * half;
    af[0] = p[0];
    af[1] = p[1];
    v2f bf;
    bf[0] = Wo[(kk * 4 + 2 * half + 0) * Ec + n];
    bf[1] = Wo[(kk * 4 + 2 * half + 1) * Ec + n];
    c = wmma4(af, bf, c);
  }
#pragma unroll
  for (int r = 0; r < 8; ++r) x[(row0 + r + 8 * half) * Ec + n] += c[r];
}

// ---------------- kernel 5: quantum FF + MLP (16 -> 64 -> 16), fused ----------
__global__ void k_ffn(float* __restrict__ x, const float* __restrict__ theta,
                      const float* __restrict__ W1, const float* __restrict__ b1,
                      const float* __restrict__ W2, const float* __restrict__ b2) {
  __shared__ float hbuf[8][16 * FFc];  // 4KB per wave, 8 waves/block
  const int wv = threadIdx.x >> 5;
  const int lane = threadIdx.x & 31;
  const int half = lane >> 4, n = lane & 15;
  const int wave = (blockIdx.x * blockDim.x + threadIdx.x) >> 5;
  const int row0 = wave << 4;
  float* hl = &hbuf[wv][0];

  // qout = cos(theta) * cos(x), directly in A layout
  v2f a[4];
#pragma unroll
  for (int kk = 0; kk < 4; ++kk) {
    const int e0 = kk * 4 + 2 * half;
    const float* p = x + (row0 + n) * Ec + e0;
    a[kk][0] = __cosf(theta[e0 + 0]) * __cosf(p[0]);
    a[kk][1] = __cosf(theta[e0 + 1]) * __cosf(p[1]);
  }

  // h = relu(qout @ W1 + b1), 4 chunks of 16 columns -> LDS (D layout)
#pragma unroll
  for (int nf = 0; nf < 4; ++nf) {
    v8f c = {};
#pragma unroll
    for (int kk = 0; kk < 4; ++kk) {
      v2f bf;
      bf[0] = W1[(kk * 4 + 2 * half + 0) * FFc + nf * 16 + n];
      bf[1] = W1[(kk * 4 + 2 * half + 1) * FFc + nf * 16 + n];
      c = wmma4(a[kk], bf, c);
    }
    float bias = b1[nf * 16 + n];
#pragma unroll
    for (int r = 0; r < 8; ++r)
      hl[(r + 8 * half) * FFc + nf * 16 + n] = fmaxf(c[r] + bias, 0.f);
  }

  // x += h @ W2 + b2  (re-load h from LDS in A layout; same-wave RAW)
  v8f c2 = {};
#pragma unroll
  for (int kk = 0; kk < 16; ++kk) {
    v2f af;
    const float* hp = hl + n * FFc + kk * 4 + 2 * half;
    af[0] = hp[0];
    af[1] = hp[1];
    v2f bf;
    bf[0] = W2[(kk * 4 + 2 * half + 0) * Ec + n];
    bf[1] = W2[(kk * 4 + 2 * half + 1) * Ec + n];
    c2 = wmma4(af, bf, c2);
  }
  float bias2 = b2[n];
#pragma unroll
  for (int r = 0; r < 8; ++r)
    x[(row0 + r + 8 * half) * Ec + n] += c2[r] + bias2;
}

// ---------------- kernel 6: LayerNorm + mean-pool + classifier ----------------
__global__ void k_final(const float* __restrict__ x, const float* __restrict__ gamma,
                        const float* __restrict__ beta, const float* __restrict__ Wc,
                        const float* __restrict__ bc, float* __restrict__ out) {
  const int b = blockIdx.x;  // one block per batch element
  __shared__ float pool[Ec];
  if (threadIdx.x < Ec) pool[threadIdx.x] = 0.f;
  __syncthreads();

  float acc[Ec];
#pragma unroll
  for (int e = 0; e < Ec; ++e) acc[e] = 0.f;

  for (int t = threadIdx.x; t < Tc; t += blockDim.x) {
    const float* row = x + (b * Tc + t) * Ec;
    float r[Ec], mu = 0.f;
#pragma unroll
    for (int e = 0; e < Ec; ++e) { r[e] = row[e]; mu += r[e]; }
    mu *= (1.f / Ec);
    float var = 0.f;
#pragma unroll
    for (int e = 0; e < Ec; ++e) { float d = r[e] - mu; var += d * d; }
    var *= (1.f / Ec);
    float inv = rsqrtf(var + 1e-5f);
#pragma unroll
    for (int e = 0; e < Ec; ++e)
      acc[e] += (r[e] - mu) * inv * gamma[e] + beta[e];
  }
#pragma unroll
  for (int e = 0; e < Ec; ++e) atomicAdd(&pool[e], acc[e]);
  __syncthreads();

  if (threadIdx.x < Cc) {
    int c = threadIdx.x;
    float s = bc[c];
#pragma unroll
    for (int e = 0; e < Ec; ++e) s += (pool[e] * (1.f / Tc)) * Wc[e * Cc + c];
    out[b * Cc + c] = s;
  }
}

// ---------------- host-side launch ----------------
extern "C" void kernel_launch(void* const* d_in, const int* in_sizes, int n_in,
                              void* d_out, int out_size, void* d_ws, size_t ws_size,
                              hipStream_t stream) {
  const int*   tokens = (const int*)d_in[0];
  const float* emb    = (const float*)d_in[1];
  const float* Wq     = (const float*)d_in[2];
  const float* Wk     = (const float*)d_in[3];
  const float* Wv     = (const float*)d_in[4];
  const float* Wo     = (const float*)d_in[5];
  const float* theta  = (const float*)d_in[6];
  const float* W1     = (const float*)d_in[7];
  const float* b1     = (const float*)d_in[8];
  const float* W2     = (const float*)d_in[9];
  const float* b2     = (const float*)d_in[10];
  const float* gamma  = (const float*)d_in[11];
  const float* beta   = (const float*)d_in[12];
  const float* Wc     = (const float*)d_in[13];
  const float* bc     = (const float*)d_in[14];
  float* out = (float*)d_out;

  float* ws = (float*)d_ws;
  float* x  = ws;            // [N,E]
  float* q  = ws + 1 * NEc;  // [N,E]
  float* k  = ws + 2 * NEc;  // [N,E]
  float* v  = ws + 3 * NEc;  // [N,E]
  float* ao = ws + 4 * NEc;  // [N,E]

  k_embed<<<NEc / 256, 256, 0, stream>>>(tokens, emb, x);
  for (int l = 0; l < Lc; ++l) {
    k_qkv<<<64, 256, 0, stream>>>(x, Wq + l * Ec * Ec, Wk + l * Ec * Ec,
                                  Wv + l * Ec * Ec, q, k, v);
    k_attn<<<64, 1024, 0, stream>>>(q, k, v, ao);  // 2048 waves, TDM-staged K/V
    k_oproj<<<64, 256, 0, stream>>>(ao, Wo + l * Ec * Ec, x);
    k_ffn<<<64, 256, 0, stream>>>(x, theta + l * Ec, W1 + l * Ec * FFc,
                                  b1 + l * FFc, W2 + l * FFc * Ec, b2 + l * Ec);
  }
  k_final<<<Bc, 256, 0, stream>>>(x, gamma, beta, Wc, bc, out);
}